// Conv2dT_7756710936583
// MI455X (gfx1250) — compile-verified
//
#include <hip/hip_runtime.h>

#define TICKS   1000
#define KH      7
#define KW      7
#define NTAPS   49
#define OUT_CH  128
#define KPAD    52      // K padded to multiple of 4
#define NPAD    1008    // ticks padded to multiple of 16

// workspace layout (float offsets)
#define BUF_OFF  0                       // [NTAPS][TICKS] tap histogram
#define F_OFF    49024                   // [KPAD][NPAD] exp-filtered input (padded with zeros)
#define ISYN_OFF (F_OFF + KPAD * NPAD)   // [OUT_CH][NPAD] synaptic current

typedef __attribute__((ext_vector_type(2))) float v2f;
typedef __attribute__((ext_vector_type(8))) float v8f;

__global__ void zero_ws_kernel(float* __restrict__ ws, int n) {
    int i = blockIdx.x * blockDim.x + threadIdx.x;
    int gs = gridDim.x * blockDim.x;
    for (; i < n; i += gs) ws[i] = 0.0f;
}

// Per-event scatter: buf[(ky*7+kx)*TICKS + tick] += value for each valid tap.
__global__ void scatter_events_kernel(const int* __restrict__ ticks,
                                      const int* __restrict__ xs,
                                      const int* __restrict__ ys,
                                      const float* __restrict__ vals,
                                      const int* __restrict__ stride_p,
                                      float* __restrict__ buf,
                                      int n_events) {
    const int s = stride_p[0];
    int i = blockIdx.x * blockDim.x + threadIdx.x;
    int gs = gridDim.x * blockDim.x;
    for (; i < n_events; i += gs) {
        const int t = ticks[i];
        const int x = xs[i];
        const int y = ys[i];
        const float v = vals[i];
        #pragma unroll
        for (int ky = 0; ky < KH; ++ky) {
            const int oy = y - ky;
            if (oy < 0 || (oy % s) != 0) continue;
            #pragma unroll
            for (int kx = 0; kx < KW; ++kx) {
                const int ox = x - kx;
                if (ox < 0 || (ox % s) != 0) continue;
                atomicAdd(&buf[(ky * KW + kx) * TICKS + t], v);
            }
        }
    }
}

// F[k,t] = decay*F[k,t-1] + buf[k,t-1]  (1-tick synaptic delay), rows k>=49 stay 0.
__global__ void exp_filter_kernel(const float* __restrict__ buf,
                                  float* __restrict__ F) {
    const int k = threadIdx.x;
    if (k >= NTAPS) return;
    const float decay = 1.0f - 0.001f / 0.01f;   // 1 - dt/tau_fall = 0.9
    float acc = 0.0f;
    for (int t = 0; t < TICKS; ++t) {
        const float in = (t >= 1) ? buf[k * TICKS + (t - 1)] : 0.0f;
        acc = acc * decay + in;                   // SCALE = 1
        F[k * NPAD + t] = acc;
    }
}

// Isyn = W (128 x 49, zero-padded K to 52) x F (52 x 1008) via v_wmma_f32_16x16x4_f32.
// Grid: exactly 8*63 = 504 waves (126 blocks x 4 waves); EXEC all-1s at every WMMA.
__global__ void __launch_bounds__(128)
wmma_gemm_kernel(const float* __restrict__ W,
                 const float* __restrict__ F,
                 float* __restrict__ Isyn) {
    const int wave = threadIdx.x >> 5;
    const int lane = threadIdx.x & 31;
    const int tile = blockIdx.x * 4 + wave;       // 0..503
    const int ntiles_n = NPAD / 16;               // 63
    const int mt = tile / ntiles_n;               // 0..7
    const int nt = tile % ntiles_n;               // 0..62

    const int half = lane >> 4;                   // 0 or 1
    const int l16  = lane & 15;
    const int m = mt * 16 + l16;                  // A row for this lane
    const int n = nt * 16 + l16;                  // B/C column for this lane

    v8f c = {};
    #pragma unroll
    for (int kk = 0; kk < KPAD; kk += 4) {
        const int kA = kk + half * 2;             // this lane holds K = kA, kA+1
        v2f a, b;
        a.x = (kA     < NTAPS) ? W[m * NTAPS + kA]     : 0.0f;
        a.y = (kA + 1 < NTAPS) ? W[m * NTAPS + kA + 1] : 0.0f;
        b.x = F[kA * NPAD + n];                   // rows >= 49 are zero-padded
        b.y = F[(kA + 1) * NPAD + n];
        c = __builtin_amdgcn_wmma_f32_16x16x4_f32(
                /*neg_a=*/false, a, /*neg_b=*/false, b,
                /*c_mod=*/(short)0, c, /*reuse_a=*/false, /*reuse_b=*/false);
    }

    // C/D layout: lane l16 = column, VGPR j = row (half*8 + j)
    #pragma unroll
    for (int j = 0; j < 8; ++j) {
        const int mrow = mt * 16 + half * 8 + j;
        Isyn[mrow * NPAD + n] = c[j];
    }
}

// Izhikevich scan: 128 channels in parallel, 1000 serial ticks.
// out[0][c][t] = spike raster, out[1][c][t] = membrane trace (post-reset).
__global__ void izhikevich_kernel(const float* __restrict__ Isyn,
                                  float* __restrict__ out) {
    const int c = threadIdx.x;                    // 0..127
    float v = -75.0f;                             // VREST
    float u = 0.0f;
    const float dt_over_C = 0.001f / 150.0f;
    for (int t = 0; t < TICKS; ++t) {
        const float I = 350.0f + Isyn[c * NPAD + t];   // I_bias=0, I_in=350
        v = v + (1.2f * (v + 75.0f) * (v + 45.0f) - u + I) * dt_over_C;
        u = u + 0.01f * (5.0f * (v + 75.0f) - u) * 0.001f;
        const bool spiked = (v >= 50.0f);
        out[c * TICKS + t] = spiked ? 1.0f : 0.0f;
        if (spiked) { v = -56.0f; u = u + 130.0f; }
        out[OUT_CH * TICKS + c * TICKS + t] = v;
    }
}

extern "C" void kernel_launch(void* const* d_in, const int* in_sizes, int n_in,
                              void* d_out, int out_size, void* d_ws, size_t ws_size,
                              hipStream_t stream) {
    const int*   ticks    = (const int*)d_in[0];
    const int*   xs       = (const int*)d_in[1];
    const int*   ys       = (const int*)d_in[2];
    const float* vals     = (const float*)d_in[3];
    const float* W        = (const float*)d_in[4];
    const int*   stride_p = (const int*)d_in[5];
    const int n_events = in_sizes[0];

    float* ws   = (float*)d_ws;
    float* buf  = ws + BUF_OFF;
    float* F    = ws + F_OFF;
    float* Isyn = ws + ISYN_OFF;
    float* out  = (float*)d_out;

    // 1) zero histogram + filtered buffer (padding must be zero for the GEMM)
    const int zero_n = F_OFF + KPAD * NPAD;
    zero_ws_kernel<<<(zero_n + 255) / 256, 256, 0, stream>>>(ws, zero_n);

    // 2) event scatter (atomic histogram, fits in L2)
    scatter_events_kernel<<<2048, 256, 0, stream>>>(ticks, xs, ys, vals,
                                                    stride_p, buf, n_events);

    // 3) exponential synapse filter (channel-independent)
    exp_filter_kernel<<<1, 64, 0, stream>>>(buf, F);

    // 4) Isyn = W x F via f32 WMMA: 8 x 63 tiles = 504 waves = 126 blocks x 4 waves
    wmma_gemm_kernel<<<126, 128, 0, stream>>>(W, F, Isyn);

    // 5) Izhikevich neuron scan -> [2, 128, 1000]
    izhikevich_kernel<<<1, 128, 0, stream>>>(Isyn, out);
}